// CameraSpline_9088150798696
// MI455X (gfx1250) — compile-verified
//
#include <hip/hip_runtime.h>

#define DEV __device__ __forceinline__

struct Quat { float w, x, y, z; };

DEV Quat qnormalize(Quat q) {
    float n2 = q.w * q.w + q.x * q.x + q.y * q.y + q.z * q.z;
    float inv = __builtin_amdgcn_rsqf(n2);   // v_rsq_f32
    return { q.w * inv, q.x * inv, q.y * inv, q.z * inv };
}

// Shortest-path slerp matching the JAX reference exactly.
DEV Quat qslerp(Quat a, Quat b, float u) {
    float d = a.w * b.w + a.x * b.x + a.y * b.y + a.z * b.z;
    d = fminf(fmaxf(d, -1.0f), 1.0f);
    float sgn = (d < 0.0f) ? -1.0f : 1.0f;
    float ad = fabsf(d);                       // in [0,1]
    float theta = acosf(ad);
    float s = __sinf(theta);                   // v_sin_f32 path
    bool ok = fabsf(s) > 1e-6f;
    float invs = __builtin_amdgcn_rcpf(ok ? s : 1.0f);  // v_rcp_f32
    float w0 = ok ? __sinf((1.0f - u) * theta) * invs : (1.0f - u);
    float w1 = (ok ? __sinf(u * theta) * invs : u) * sgn;
    return { w0 * a.w + w1 * b.w,
             w0 * a.x + w1 * b.x,
             w0 * a.y + w1 * b.y,
             w0 * a.z + w1 * b.z };
}

DEV Quat load_quat(const float4* __restrict__ q, int idx) {
    float4 v = q[idx];                 // global_load_b128 (16B-aligned gather)
    return { v.x, v.y, v.z, v.w };     // storage order is (w,x,y,z)
}

__global__ __launch_bounds__(256) void spline_pose_kernel(
    const float*  __restrict__ t,
    const float*  __restrict__ ctrl_trans,
    const float4* __restrict__ ctrl_quats,
    const int*    __restrict__ Nptr,
    float* __restrict__ outR,
    float* __restrict__ outT,
    int M, int K)
{
    int m = blockIdx.x * blockDim.x + threadIdx.x;
    if (m >= M) return;

    // Prefetch the linear t stream ahead (global_prefetch_b8).
    if (m + 8192 < M) __builtin_prefetch(&t[m + 8192], 0, 0);

    int nm1 = *Nptr - 1;               // uniform -> scalar load
    if (nm1 < 1) nm1 = 1;
    float kscale = (float)(K - 1) / (float)nm1;

    float tv = __builtin_nontemporal_load(&t[m]);
    float t_ctrl = tv * kscale;
    int i = (int)floorf(t_ctrl);
    i = min(max(i, 0), K - 2);
    float u = t_ctrl - (float)i;

    int im1 = max(i - 1, 0);
    int ip1 = i + 1;
    int ip2 = min(i + 2, K - 1);

    // ---------------- translation: cubic Hermite ----------------
    const float* pm = ctrl_trans + 3 * (size_t)im1;
    const float* p0 = ctrl_trans + 3 * (size_t)i;
    const float* p1 = ctrl_trans + 3 * (size_t)ip1;
    const float* p2 = ctrl_trans + 3 * (size_t)ip2;

    float p0x = p0[0], p0y = p0[1], p0z = p0[2];
    float p1x = p1[0], p1y = p1[1], p1z = p1[2];
    float pmx = pm[0], pmy = pm[1], pmz = pm[2];
    float p2x = p2[0], p2y = p2[1], p2z = p2[2];

    float m0x, m0y, m0z, m1x, m1y, m1z;
    if (i > 0) {
        m0x = 0.5f * (p1x - pmx); m0y = 0.5f * (p1y - pmy); m0z = 0.5f * (p1z - pmz);
    } else {
        m0x = p1x - p0x; m0y = p1y - p0y; m0z = p1z - p0z;
    }
    if (ip1 < K - 1) {
        m1x = 0.5f * (p2x - p0x); m1y = 0.5f * (p2y - p0y); m1z = 0.5f * (p2z - p0z);
    } else {
        m1x = p1x - p0x; m1y = p1y - p0y; m1z = p1z - p0z;
    }

    float u2 = u * u, u3 = u2 * u;
    float h00 =  2.0f * u3 - 3.0f * u2 + 1.0f;
    float h10 =         u3 - 2.0f * u2 + u;
    float h01 = -2.0f * u3 + 3.0f * u2;
    float h11 =         u3 -        u2;

    float Tx = h00 * p0x + h10 * m0x + h01 * p1x + h11 * m1x;
    float Ty = h00 * p0y + h10 * m0y + h01 * p1y + h11 * m1y;
    float Tz = h00 * p0z + h10 * m0z + h01 * p1z + h11 * m1z;

    // ---------------- rotation: squad (double slerp) ----------------
    Quat q_i  = qnormalize(load_quat(ctrl_quats, i));
    Quat q_i1 = qnormalize(load_quat(ctrl_quats, ip1));
    Quat q_m  = qnormalize(load_quat(ctrl_quats, im1));
    Quat q_2  = qnormalize(load_quat(ctrl_quats, ip2));

    Quat s_main = qslerp(q_i, q_i1, u);
    Quat s_aux  = qslerp(q_m, q_2, u);
    float u_squad = 2.0f * u * (1.0f - u);
    Quat q = qnormalize(qslerp(s_main, s_aux, u_squad));

    float w = q.w, x = q.x, y = q.y, z = q.z;
    float r00 = 1.0f - 2.0f * (y * y + z * z);
    float r01 = 2.0f * (x * y - w * z);
    float r02 = 2.0f * (x * z + w * y);
    float r10 = 2.0f * (x * y + w * z);
    float r11 = 1.0f - 2.0f * (x * x + z * z);
    float r12 = 2.0f * (y * z - w * x);
    float r20 = 2.0f * (x * z - w * y);
    float r21 = 2.0f * (y * z + w * x);
    float r22 = 1.0f - 2.0f * (x * x + y * y);

    // Write-once 96MB output stream: non-temporal so it doesn't evict the
    // ~11MB control tables (which we want resident in the 192MB L2).
    float* __restrict__ rbase = outR + (size_t)m * 9;
    __builtin_nontemporal_store(r00, rbase + 0);
    __builtin_nontemporal_store(r01, rbase + 1);
    __builtin_nontemporal_store(r02, rbase + 2);
    __builtin_nontemporal_store(r10, rbase + 3);
    __builtin_nontemporal_store(r11, rbase + 4);
    __builtin_nontemporal_store(r12, rbase + 5);
    __builtin_nontemporal_store(r20, rbase + 6);
    __builtin_nontemporal_store(r21, rbase + 7);
    __builtin_nontemporal_store(r22, rbase + 8);

    float* __restrict__ tbase = outT + (size_t)m * 3;
    __builtin_nontemporal_store(Tx, tbase + 0);
    __builtin_nontemporal_store(Ty, tbase + 1);
    __builtin_nontemporal_store(Tz, tbase + 2);
}

extern "C" void kernel_launch(void* const* d_in, const int* in_sizes, int n_in,
                              void* d_out, int out_size, void* d_ws, size_t ws_size,
                              hipStream_t stream) {
    const float*  t          = (const float*)d_in[0];
    const float*  ctrl_trans = (const float*)d_in[1];
    const float4* ctrl_quats = (const float4*)d_in[2];
    const int*    Nptr       = (const int*)d_in[3];

    int M = in_sizes[0];
    int K = in_sizes[1] / 3;

    float* outR = (float*)d_out;              // [M,3,3] flattened first
    float* outT = outR + (size_t)M * 9;       // then [M,3]

    const int block = 256;                    // 8 wave32s per block
    const int grid  = (M + block - 1) / block;
    hipLaunchKernelGGL(spline_pose_kernel, dim3(grid), dim3(block), 0, stream,
                       t, ctrl_trans, ctrl_quats, Nptr, outR, outT, M, K);
}